// GeoRelationship_16836271800565
// MI455X (gfx1250) — compile-verified
//
#include <hip/hip_runtime.h>

typedef __attribute__((ext_vector_type(2))) float v2f;
typedef __attribute__((ext_vector_type(8))) float v8f;

#define HDIM 768
#define LN_EPS 1e-5f

// LDS: W pre-packed in B-operand order.
// For k-triplet t (K base = 4t), column c, lane-half h:
//   sWv[(t*HDIM + c)*2 + h] = { W[4t+2h][c], W[4t+2h+1][c] }  (K >= 11 zero-padded)
// so each WMMA B operand is ONE contiguous ds_load_b64 per lane.

__global__ __launch_bounds__(256) void georel_wmma_kernel(
    const float* __restrict__ target,   // 4
    const float* __restrict__ boxes,    // N x 4
    const float* __restrict__ Wm,       // 11 x 768 row-major
    const float* __restrict__ bias,     // 768
    const float* __restrict__ gamma,    // 768
    const float* __restrict__ beta,     // 768
    float* __restrict__ out,            // N x 768
    int N)
{
    __shared__ v2f   sWv[3 * HDIM * 2];   // 36 KB
    __shared__ float sB[HDIM];
    __shared__ float sG[HDIM];
    __shared__ float sBe[HDIM];

    const int tid = threadIdx.x;
    for (int i = tid; i < 3 * HDIM * 2; i += 256) {
        const int t   = i / (HDIM * 2);
        const int rem = i - t * (HDIM * 2);
        const int col = rem >> 1;
        const int h   = rem & 1;
        const int kb  = t * 4 + h * 2;
        v2f val;
        val.x = (kb     < 11) ? Wm[kb * HDIM + col]       : 0.0f;
        val.y = (kb + 1 < 11) ? Wm[(kb + 1) * HDIM + col] : 0.0f;
        sWv[i] = val;
    }
    for (int i = tid; i < HDIM; i += 256) {
        sB[i]  = bias[i];
        sG[i]  = gamma[i];
        sBe[i] = beta[i];
    }
    __syncthreads();

    const int lane = tid & 31;
    const int wave = tid >> 5;
    const int half = lane >> 4;     // 0 = lanes 0-15, 1 = lanes 16-31
    const int lcol = lane & 15;

    const int r0 = blockIdx.x * 128 + wave * 16;   // first row of this wave's 16-row tile
    const int r  = r0 + lcol;
    const int rr = (r < N) ? r : (N - 1);          // clamp for tail loads

    // ---- geometric features for row (lane & 15) ----
    const float tx0 = target[0], ty0 = target[1], tx1 = target[2], ty1 = target[3];
    const float4 bx = ((const float4*)boxes)[rr];
    const float x0 = bx.x, y0 = bx.y, x1 = bx.z, y1 = bx.w;

    const float w1 = tx1 - tx0, h1 = ty1 - ty0;
    const float w2 = x1 - x0,   h2 = y1 - y0;

    const float c1x = (tx0 + tx1) * 0.5f, c1y = (ty0 + ty1) * 0.5f;
    const float c2x = (x0 + x1) * 0.5f,   c2y = (y0 + y1) * 0.5f;
    const float dx = c2x - c1x, dy = c2y - c1y;
    const float dist = sqrtf(dx * dx + dy * dy);

    const bool overlap = (tx0 < x1) && (tx1 > x0) && (ty0 < y1) && (ty1 > y0);
    const float xi0 = fmaxf(tx0, x0), yi0 = fmaxf(ty0, y0);
    const float xi1 = fminf(tx1, x1), yi1 = fminf(ty1, y1);
    const float inter = overlap ? (xi1 - xi0) * (yi1 - yi0) : 0.0f;
    const float a1 = w1 * h1, a2 = w2 * h2;
    const float uni = a1 + a2 - inter;

    // degree = atan2(|dy|,|dx|) in [0,90]; tan(22.5)=.41421356, tan(67.5)=2.41421356
    const float ady = fabsf(dy), adx = fabsf(dx);
    const float bucket = (ady > adx * 2.41421356f) ? 2.0f
                       : (ady > adx * 0.41421356f) ? 1.0f : 8.0f;

    const float f0 = w1, f1 = h1, f2 = w2, f3 = h2;
    const float f4 = fabsf(w1 - w2), f5 = fabsf(h1 - h2);
    const float f6 = dist;
    const float f7 = inter / uni, f8 = inter / a1, f9 = inter / a2;
    const float f10 = bucket;

    // ---- A tiles (16x4 f32): lanes 0-15 hold K=k,k+1; lanes 16-31 hold K=k+2,k+3 ----
    v2f A0, A1, A2;
    A0.x = half ? f2  : f0;  A0.y = half ? f3 : f1;   // K 0..3
    A1.x = half ? f6  : f4;  A1.y = half ? f7 : f5;   // K 4..7
    A2.x = half ? f10 : f8;  A2.y = half ? 0.0f : f9; // K 8..11 (11 padded)

    // ---- pass 1: per-row sum / sumsq from WMMA accumulators ----
    float rs[8], rq[8];
#pragma unroll
    for (int v = 0; v < 8; ++v) { rs[v] = 0.0f; rq[v] = 0.0f; }

    for (int jt = 0; jt < HDIM / 16; ++jt) {
        const int col = jt * 16 + lcol;
        const int ci  = col * 2 + half;
        const v2f B0 = sWv[ci];
        const v2f B1 = sWv[HDIM * 2 + ci];
        const v2f B2 = sWv[HDIM * 4 + ci];

        v8f c = {};
        c = __builtin_amdgcn_wmma_f32_16x16x4_f32(false, A0, false, B0, (short)0, c, false, false);
        c = __builtin_amdgcn_wmma_f32_16x16x4_f32(false, A1, false, B1, (short)0, c, false, false);
        c = __builtin_amdgcn_wmma_f32_16x16x4_f32(false, A2, false, B2, (short)0, c, false, false);

        const float bb = sB[col];
#pragma unroll
        for (int v = 0; v < 8; ++v) {
            const float x = c[v] + bb;
            rs[v] += x;
            rq[v] += x * x;
        }
    }

    // reduce across the 16 columns held by each lane-half (masks 1..8 stay in-half)
#pragma unroll
    for (int v = 0; v < 8; ++v) {
        float s = rs[v], q = rq[v];
#pragma unroll
        for (int m = 1; m < 16; m <<= 1) {
            s += __shfl_xor(s, m, 32);
            q += __shfl_xor(q, m, 32);
        }
        const float mean = s * (1.0f / HDIM);
        const float var  = q * (1.0f / HDIM) - mean * mean;
        rs[v] = mean;
        rq[v] = rsqrtf(var + LN_EPS);
    }

    // ---- pass 2: recompute tiles, normalize, ReLU, store ----
    const bool fullTile = (r0 + 16 <= N);   // wave-uniform: no tail guard needed

    for (int jt = 0; jt < HDIM / 16; ++jt) {
        const int col = jt * 16 + lcol;
        const int ci  = col * 2 + half;
        const v2f B0 = sWv[ci];
        const v2f B1 = sWv[HDIM * 2 + ci];
        const v2f B2 = sWv[HDIM * 4 + ci];

        v8f c = {};
        c = __builtin_amdgcn_wmma_f32_16x16x4_f32(false, A0, false, B0, (short)0, c, false, false);
        c = __builtin_amdgcn_wmma_f32_16x16x4_f32(false, A1, false, B1, (short)0, c, false, false);
        c = __builtin_amdgcn_wmma_f32_16x16x4_f32(false, A2, false, B2, (short)0, c, false, false);

        const float bb = sB[col], gg = sG[col], be = sBe[col];
        float* outp = out + (size_t)(r0 + half * 8) * HDIM + col;

        if (fullTile) {
#pragma unroll
            for (int v = 0; v < 8; ++v) {
                const float x = c[v] + bb;
                const float y = (x - rs[v]) * rq[v] * gg + be;
                outp[(size_t)v * HDIM] = fmaxf(y, 0.0f);
            }
        } else {
#pragma unroll
            for (int v = 0; v < 8; ++v) {
                const int row = r0 + v + half * 8;
                if (row < N) {
                    const float x = c[v] + bb;
                    const float y = (x - rs[v]) * rq[v] * gg + be;
                    out[(size_t)row * HDIM + col] = fmaxf(y, 0.0f);
                }
            }
        }
    }
}

extern "C" void kernel_launch(void* const* d_in, const int* in_sizes, int n_in,
                              void* d_out, int out_size, void* d_ws, size_t ws_size,
                              hipStream_t stream) {
    const float* target = (const float*)d_in[0];
    const float* boxes  = (const float*)d_in[1];
    const float* Wm     = (const float*)d_in[2];
    const float* bias   = (const float*)d_in[3];
    const float* gamma  = (const float*)d_in[4];
    const float* beta   = (const float*)d_in[5];
    float* out = (float*)d_out;

    const int N = in_sizes[1] / 4;          // ocr_boxes is N x 4
    const int blocks = (N + 127) / 128;     // 128 rows per block (8 waves x 16 rows)

    georel_wmma_kernel<<<blocks, 256, 0, stream>>>(target, boxes, Wm, bias, gamma, beta, out, N);
}